// AttentionGenerator_51694226375000
// MI455X (gfx1250) — compile-verified
//
#include <hip/hip_runtime.h>
#include <hip/hip_bf16.h>

#define ALPHA 0.1f

typedef float v2f __attribute__((ext_vector_type(2)));
typedef float v8f __attribute__((ext_vector_type(8)));

// ---------------------------------------------------------------------------
// Kernel 1: fold Wf/bf into u-vectors.
//   u1[c] = sum_o Wa[o]    * Wf[o,c]   (c = 0..255)
//   u2[c] = sum_o Wa[64+o] * Wf[o,c]
//   c1    = Wa[:64]  . bf
//   c2    = Wa[64:]  . bf
// ws layout (floats): [0,256)=u1  [256,512)=u2  [512]=c1  [513]=c2
// ---------------------------------------------------------------------------
__global__ void __launch_bounds__(256)
attn_fold_u(const float* __restrict__ Wf, const float* __restrict__ bf,
            const float* __restrict__ Wa, float* __restrict__ uws) {
    const int c = threadIdx.x;              // 0..255
    float u1 = 0.f, u2 = 0.f;
#pragma unroll 8
    for (int o = 0; o < 64; ++o) {
        const float w = Wf[o * 256 + c];
        u1 += Wa[o] * w;
        u2 += Wa[64 + o] * w;
    }
    uws[c]       = u1;
    uws[256 + c] = u2;
    if (c < 2) {
        float s = 0.f;
        for (int o = 0; o < 64; ++o) s += Wa[c * 64 + o] * bf[o];
        uws[512 + c] = s;
    }
}

// ---------------------------------------------------------------------------
// Kernel 2: HBM-bound streaming reduction over channels via fp32 WMMA.
//   s1[b,n] = u1 . feat[b,:,n] + c1   (n = t*18+v, 0..4607)
//   s2[b,n] = u2 . feat[b,:,n] + c2
// One wave owns a 16-position tile; 64 chained V_WMMA_F32_16X16X4_F32 steps
// sweep the 256 channels. A-matrix rows 2..15 are forced to zero with
// branchless selects so EXEC stays all-ones for every WMMA.
// ---------------------------------------------------------------------------
__global__ void __launch_bounds__(256)
attn_reduce_s(const float* __restrict__ feat, const float* __restrict__ uws,
              float* __restrict__ s1, float* __restrict__ s2) {
    const int lane  = threadIdx.x & 31;
    const int wave  = threadIdx.x >> 5;
    const int b     = blockIdx.x;                 // 0..31
    const int tile  = blockIdx.y * 8 + wave;      // 0..287
    const int n0    = tile * 16;
    const int l15   = lane & 15;
    const int khalf = (lane >> 4) << 1;           // 0 for lanes 0-15, 2 for 16-31

    // B-matrix feed: VGPR0 <- feat[ch = c0+khalf][n0+l15], VGPR1 <- ch+1
    const float* fptr = feat + ((size_t)b * 256 + (size_t)khalf) * 4608 + n0 + l15;
    // A-matrix feed: row m=l15 uses u1 (m==0) / u2 (m==1) / 0 (m>=2)
    const float* uptr = uws + (l15 == 1 ? 256 : 0) + khalf;
    const bool   aval = (l15 < 2);

    v8f acc = {};
#pragma unroll 4
    for (int c0 = 0; c0 < 256; c0 += 4) {
        v2f bm, am;
        bm.x = fptr[(size_t)c0 * 4608];
        bm.y = fptr[(size_t)(c0 + 1) * 4608];
        const float ua = uptr[c0];
        const float ub = uptr[c0 + 1];
        am.x = aval ? ua : 0.0f;    // v_cndmask: EXEC stays all-1s
        am.y = aval ? ub : 0.0f;
        acc = __builtin_amdgcn_wmma_f32_16x16x4_f32(
            /*neg_a=*/false, am, /*neg_b=*/false, bm,
            /*c_mod=*/(short)0, acc, /*reuse_a=*/false, /*reuse_b=*/false);
    }

    // D layout: VGPR0 lanes 0-15 = row M=0 (s1), VGPR1 lanes 0-15 = row M=1 (s2)
    if (lane < 16) {
        const float c1  = uws[512];
        const float c2  = uws[513];
        const int   idx = b * 4608 + n0 + lane;
        s1[idx] = acc[0] + c1;
        s2[idx] = acc[1] + c2;
    }
}

// ---------------------------------------------------------------------------
// Kernel 3: attention rows. One thread per (b,t,i):
//   score[j] = s1[b,t,i] + s2[b,t,j] + ba ; leaky-ReLU ; exp * A[i,j] ;
//   softmax over j. 147456 threads total = 576 blocks x 256.
// ---------------------------------------------------------------------------
__global__ void __launch_bounds__(256)
attn_softmax(const float* __restrict__ s1, const float* __restrict__ s2,
             const float* __restrict__ A, const float* __restrict__ ba_p,
             float* __restrict__ out) {
    const int r    = blockIdx.x * 256 + threadIdx.x;  // 0..147455 == bt*18 + i
    const int i    = r % 18;
    const int base = r - i;                           // bt*18
    const float ba = ba_p[0];
    const float si = s1[r];
    const float* Arow = A  + i * 18;
    const float* s2r  = s2 + base;

    float ex[18];
    float sum = 0.f;
#pragma unroll
    for (int j = 0; j < 18; ++j) {
        const float sc = si + s2r[j] + ba;
        const float e  = sc > 0.f ? sc : ALPHA * sc;
        const float v  = expf(e) * Arow[j];
        ex[j] = v;
        sum  += v;
    }
    const float inv = 1.0f / sum;
    float* o = out + (size_t)r * 18;
#pragma unroll
    for (int j = 0; j < 18; ++j) o[j] = ex[j] * inv;
}

// ---------------------------------------------------------------------------
extern "C" void kernel_launch(void* const* d_in, const int* in_sizes, int n_in,
                              void* d_out, int out_size, void* d_ws, size_t ws_size,
                              hipStream_t stream) {
    const float* feat = (const float*)d_in[0];   // (32,256,256,18)
    const float* A    = (const float*)d_in[1];   // (18,18)
    const float* Wf   = (const float*)d_in[2];   // (64,256)
    const float* bf   = (const float*)d_in[3];   // (64,)
    const float* Wa   = (const float*)d_in[4];   // (128,)
    const float* ba   = (const float*)d_in[5];   // (1,)
    float* out = (float*)d_out;                  // (32,256,18,18)

    float* ws = (float*)d_ws;
    float* s1 = ws + 1024;                       // 32*4608 floats
    float* s2 = s1 + 32 * 4608;                  // 32*4608 floats

    attn_fold_u <<<1, 256, 0, stream>>>(Wf, bf, Wa, ws);
    attn_reduce_s<<<dim3(32, 36), 256, 0, stream>>>(feat, ws, s1, s2);
    attn_softmax <<<576, 256, 0, stream>>>(s1, s2, A, ba, out);
}